// E66LowRankHCell_4887672783149
// MI455X (gfx1250) — compile-verified
//
#include <hip/hip_runtime.h>
#include <hip/hip_bf16.h>

typedef float v2f __attribute__((ext_vector_type(2)));
typedef float v8f __attribute__((ext_vector_type(8)));

#define TT   1024
#define BB   32
#define DD   1024
#define RR   256
#define NROW (TT * BB)   // 32768 rows of x flattened
#define SLAB (BB * DD)   // 32768 elements per timestep slab
#define NBLK 16          // persistent scan workgroups (cluster-sized)

#if __has_builtin(__builtin_amdgcn_tensor_load_to_lds) && \
    __has_builtin(__builtin_amdgcn_s_wait_tensorcnt)
#define USE_TDM 1
#else
#define USE_TDM 0
#endif

// arity of the TDM builtin differs between toolchains; the gfx1250 TDM header
// ships only with the toolchain whose builtin takes 6 args.
#if __has_include(<hip/amd_detail/amd_gfx1250_TDM.h>)
#define TDM_SIX_ARGS 1
#else
#define TDM_SIX_ARGS 0
#endif

__device__ __forceinline__ v8f wmma_f32(v2f a, v2f b, v8f c) {
  // D = A(16x4,f32) * B(4x16,f32) + C(16x16,f32), fp32 matrix pipe
  return __builtin_amdgcn_wmma_f32_16x16x4_f32(false, a, false, b, (short)0, c,
                                               false, false);
}

__device__ __forceinline__ float2 ld2(const float* p) {
  return *(const float2*)p;
}

#if USE_TDM
typedef unsigned u32x4 __attribute__((ext_vector_type(4)));
typedef int i32x8 __attribute__((ext_vector_type(8)));
typedef int i32x4 __attribute__((ext_vector_type(4)));

// LDS aperture keeps the byte offset in addr[31:0] (ISA 10.2 aperture calc),
// so truncating the flat address of a __shared__ object gives the D# lds_addr.
__device__ __forceinline__ unsigned lds_byte_addr(const void* p) {
  return (unsigned)(unsigned long long)p;
}

// TDM: load a 16-row x 32-col fp32 tile of x (row-major, row stride DD) into
// LDS with 2-DWORD padding after every 32 DWORDs -> 34-float LDS row stride
// (bank-conflict-free for the 16x16x4 A-tile ds_load_b64 pattern).
__device__ __forceinline__ void tdm_load_xtile(const float* x, int row0, int kc,
                                               unsigned lds_addr) {
  unsigned long long ga =
      (unsigned long long)(const void*)(x + (size_t)row0 * DD + kc);
  u32x4 g0;
  g0.x = 1u;                                    // count=1 (valid user D#)
  g0.y = lds_addr;                              // D# lds_addr (bytes)
  g0.z = (unsigned)(ga & 0xffffffffull);        // global_addr[31:0]
  g0.w = (unsigned)((ga >> 32) & 0x1ffffffull)  // global_addr[56:32]
         | (2u << 30);                          // type = 2 ("image")
  i32x8 g1;
  g1[0] = (int)((2u << 16)    // data_size = 4 bytes
                | (1u << 20)  // pad_enable
                | (4u << 22)  // pad_interval: 32 DWORDs
                | (1u << 25)); // pad_amount: 2 DWORDs
  g1[1] = (int)((unsigned)DD << 16);                  // tensor_dim0 = 1024
  g1[2] = (int)(((unsigned)NROW & 0xffffu) << 16);    // tensor_dim1 = 32768
  g1[3] = (int)(32u << 16);                           // tile_dim0 = 32
  g1[4] = (int)16u;                                   // tile_dim1=16, dim2=0
  g1[5] = (int)(unsigned)DD;                          // tensor_dim0_stride
  g1[6] = 0;
  g1[7] = 0;
  i32x4 gz4 = {0, 0, 0, 0};                           // groups 2/3 unused (2D)
#if TDM_SIX_ARGS
  i32x8 gz8 = {0, 0, 0, 0, 0, 0, 0, 0};
  __builtin_amdgcn_tensor_load_to_lds(g0, g1, gz4, gz4, gz8, 0);
#else
  __builtin_amdgcn_tensor_load_to_lds(g0, g1, gz4, gz4, 0);
#endif
}
#endif

// ---------------------------------------------------------------------------
// Kernel 1: fused dual projection  ax = x@Wa^T + ba,  wx = x@Wx^T + bx
// ax[t] is stashed into the h-output slab at index t+1 (read-then-overwritten
// by the scan), wx[t] is stashed into the out slab at index t.
// Block = 256 thr (8 waves); tile = 16 rows x 128 cols (16 cols per wave).
// X K-chunks are DMA'd by the Tensor Data Mover, double-buffered on TENSORcnt.
// ---------------------------------------------------------------------------
__global__ __launch_bounds__(256)
void proj_kernel(const float* __restrict__ x,
                 const float* __restrict__ Wa, const float* __restrict__ ba,
                 const float* __restrict__ Wx, const float* __restrict__ bx,
                 float* __restrict__ outbuf, float* __restrict__ hbase) {
  __shared__ float Xs[2][16][34];              // padded rows -> no bank conflicts
  const int tid  = threadIdx.x;
  const int wave = tid >> 5, lane = tid & 31;
  const int row0 = blockIdx.y * 16;
  const int n0   = blockIdx.x * 128 + wave * 16;
  const int l15  = lane & 15;
  const int kb   = (lane >> 4) * 2;            // K-pair select per A/B layout

  v8f accA = {}; v8f accX = {};

  auto consume = [&](int buf, int kc) {
#pragma unroll
    for (int k4 = 0; k4 < 32; k4 += 4) {
      float2 av = ld2(&Xs[buf][l15][k4 + kb]); // ds_load_b64
      v2f a; a.x = av.x; a.y = av.y;
      float2 bva = ld2(Wa + (n0 + l15) * DD + kc + k4 + kb);
      float2 bvx = ld2(Wx + (n0 + l15) * DD + kc + k4 + kb);
      v2f bA; bA.x = bva.x; bA.y = bva.y;
      v2f bX; bX.x = bvx.x; bX.y = bvx.y;
      accA = wmma_f32(a, bA, accA);            // reuse A tile for both weights
      accX = wmma_f32(a, bX, accX);
    }
  };

#if USE_TDM
  if (wave == 0) tdm_load_xtile(x, row0, 0, lds_byte_addr(&Xs[0][0][0]));
  int buf = 0;
  for (int kc = 0; kc < DD; kc += 32) {
    if (wave == 0) {
      if (kc + 32 < DD) {
        tdm_load_xtile(x, row0, kc + 32, lds_byte_addr(&Xs[buf ^ 1][0][0]));
        __builtin_amdgcn_s_wait_tensorcnt(1);  // in-order: chunk 'buf' landed
      } else {
        __builtin_amdgcn_s_wait_tensorcnt(0);  // last chunk landed
      }
    }
    __syncthreads();                           // publish LDS to all waves
    consume(buf, kc);
    __syncthreads();                           // chunk fully consumed
    buf ^= 1;
  }
#else
  const int lm = tid >> 4;                     // staging row 0..15
  const int lk = (tid & 15) * 2;               // staging col pair
  for (int kc = 0; kc < DD; kc += 32) {
    float2 xv = ld2(x + (row0 + lm) * DD + kc + lk);
    Xs[0][lm][lk] = xv.x; Xs[0][lm][lk + 1] = xv.y;
    __syncthreads();
    consume(0, kc);
    __syncthreads();
  }
#endif

  const float biasA = ba[n0 + l15];
  const float biasX = bx[n0 + l15];
  const int rb = row0 + ((lane >> 4) << 3);
#pragma unroll
  for (int v = 0; v < 8; ++v) {
    int r = rb + v;
    int c = n0 + l15;
    hbase[(r + BB) * DD + c] = accA[v] + biasA;  // ax[t] -> h slab t+1
    outbuf[r * DD + c]       = accX[v] + biasX;  // wx[t] -> out slab t
  }
}

// ---------------------------------------------------------------------------
// Init: zero soft-barrier counter, copy h0 into h[0]
// ---------------------------------------------------------------------------
__global__ void init_kernel(const float* __restrict__ h0,
                            float* __restrict__ hbase,
                            unsigned* __restrict__ bar) {
  int i = blockIdx.x * blockDim.x + threadIdx.x;
  if (i < 64) bar[i] = 0u;
  if (i < SLAB) hbase[i] = h0[i];
}

// ---------------------------------------------------------------------------
// Device-wide soft barrier for the 16 persistent workgroups.
// Monotonic counter (reset by init_kernel each launch -> graph-replay safe).
// Also signals the CDNA5 cluster barrier (S_NOP when not cluster-dispatched).
// ---------------------------------------------------------------------------
__device__ __forceinline__ void gbar(unsigned* bar, unsigned target) {
  __threadfence();
  __syncthreads();
  if (threadIdx.x == 0) {
    atomicAdd(bar, 1u);
    while (atomicAdd(bar, 0u) < target) { __builtin_amdgcn_s_sleep(2); }
    __builtin_amdgcn_s_cluster_barrier();
  }
  __syncthreads();
  __threadfence();
}

// ---------------------------------------------------------------------------
// Kernel 2: persistent recurrence. 16 blocks x 256 thr, loops t = 0..T-1.
// Phase 1: p = h[t] @ V^T   (each block owns a 16-col slice of R=256;
//          K=1024 split across 8 waves, partials reduced in LDS)
// Phase 2: v = tanh(p @ U^T + wx); h' = a*h + (1-a)*v; out = h'*silu(h')
//          (each block owns 64 cols of D; one 16x16 tile per wave)
// ---------------------------------------------------------------------------
__global__ __launch_bounds__(256)
void scan_kernel(const float* __restrict__ U, const float* __restrict__ V,
                 float* __restrict__ outbuf, float* __restrict__ hbase,
                 float* __restrict__ pbuf, unsigned* __restrict__ bar) {
  __shared__ float red[8][512];                // 8 wave-partials of 32x16
  const int g    = blockIdx.x;                 // 0..15
  const int tid  = threadIdx.x;
  const int wave = tid >> 5, lane = tid & 31;
  const int l15  = lane & 15;
  const int kb   = (lane >> 4) * 2;
  const int k0   = wave * 128;                 // K slice for phase 1
  const int pcol = g * 16 + l15;               // V row / p column
  const int mt   = wave >> 2, nt = wave & 3;   // phase-2 tile coords
  const int e0   = g * 64 + nt * 16;           // phase-2 output column base

  for (int t = 0; t < TT; ++t) {
    const float* hprev = hbase + (size_t)t * SLAB;
    float* hnext = hbase + (size_t)(t + 1) * SLAB;
    float* outt  = outbuf + (size_t)t * SLAB;

    // warm the lines phase 2 will need
    __builtin_prefetch(outt + (mt * 16) * DD + e0, 0, 1);
    __builtin_prefetch(hnext + (mt * 16) * DD + e0, 0, 1);

    // ---- phase 1: K-split partial p tiles (rows 0-15 and 16-31) ----
    v8f acc0 = {}; v8f acc1 = {};
    const float* vrow = V + (size_t)pcol * DD;
#pragma unroll 4
    for (int k = k0; k < k0 + 128; k += 4) {
      float2 bv  = ld2(vrow + k + kb);
      float2 a0v = ld2(hprev + l15 * DD + k + kb);
      float2 a1v = ld2(hprev + (16 + l15) * DD + k + kb);
      v2f b;  b.x  = bv.x;  b.y  = bv.y;
      v2f a0; a0.x = a0v.x; a0.y = a0v.y;
      v2f a1; a1.x = a1v.x; a1.y = a1v.y;
      acc0 = wmma_f32(a0, b, acc0);
      acc1 = wmma_f32(a1, b, acc1);
    }
    {
      const int rb = (lane >> 4) << 3;
#pragma unroll
      for (int v = 0; v < 8; ++v) {
        red[wave][(rb + v) * 16 + l15]      = acc0[v];
        red[wave][(16 + rb + v) * 16 + l15] = acc1[v];
      }
    }
    __syncthreads();
#pragma unroll
    for (int e = tid; e < 512; e += 256) {     // LDS tree-reduce 8 partials
      float s = 0.f;
#pragma unroll
      for (int w = 0; w < 8; ++w) s += red[w][e];
      pbuf[(e >> 4) * RR + g * 16 + (e & 15)] = s;
    }
    gbar(bar, (unsigned)(NBLK * (2 * t + 1))); // p complete device-wide

    // ---- phase 2: v = tanh(p @ U^T + wx) + gated update + silu out ----
    v8f acc = {};
    const float* urow = U + (size_t)(e0 + l15) * RR;
    const float* prow = pbuf + (mt * 16 + l15) * RR;
#pragma unroll 4
    for (int k = 0; k < RR; k += 4) {
      float2 av = ld2(prow + k + kb);
      float2 bv = ld2(urow + k + kb);
      v2f a; a.x = av.x; a.y = av.y;
      v2f b; b.x = bv.x; b.y = bv.y;
      acc = wmma_f32(a, b, acc);
    }
    const int rb = mt * 16 + ((lane >> 4) << 3);
#pragma unroll
    for (int v = 0; v < 8; ++v) {
      int idx = (rb + v) * DD + e0 + l15;
      float wx = outt[idx];                    // stashed x-projection
      float ax = hnext[idx];                   // stashed alpha-projection
      float hp = hprev[idx];
      float alpha = __builtin_amdgcn_rcpf(1.f + __expf(-ax));
      float z  = acc[v] + wx;
      float e2 = __expf(-2.f * fabsf(z));
      float th = copysignf((1.f - e2) * __builtin_amdgcn_rcpf(1.f + e2), z);
      float hn = alpha * hp + (1.f - alpha) * th;
      float sg = __builtin_amdgcn_rcpf(1.f + __expf(-hn));
      hnext[idx] = hn;                         // h[t+1]
      outt[idx]  = hn * hn * sg;               // h' * silu(h')
    }
    gbar(bar, (unsigned)(NBLK * (2 * t + 2))); // h[t+1] complete device-wide
  }
}

// ---------------------------------------------------------------------------
extern "C" void kernel_launch(void* const* d_in, const int* in_sizes, int n_in,
                              void* d_out, int out_size, void* d_ws,
                              size_t ws_size, hipStream_t stream) {
  (void)in_sizes; (void)n_in; (void)out_size; (void)ws_size;
  const float* x  = (const float*)d_in[0];
  const float* h0 = (const float*)d_in[1];
  const float* Wa = (const float*)d_in[2];
  const float* ba = (const float*)d_in[3];
  const float* U  = (const float*)d_in[4];
  const float* V  = (const float*)d_in[5];
  const float* Wx = (const float*)d_in[6];
  const float* bx = (const float*)d_in[7];

  float* outbuf = (float*)d_out;                    // [T,B,D] output
  float* hbase  = outbuf + (size_t)TT * SLAB;       // [T+1,B,D] h sequence
  unsigned* bar = (unsigned*)d_ws;                  // soft-barrier counter
  float* pbuf   = (float*)d_ws + 64;                // [B,R] scratch (32 KB)

  init_kernel<<<dim3((SLAB + 255) / 256), 256, 0, stream>>>(h0, hbase, bar);
  proj_kernel<<<dim3(DD / 128, NROW / 16), 256, 0, stream>>>(
      x, Wa, ba, Wx, bx, outbuf, hbase);
  scan_kernel<<<dim3(NBLK), 256, 0, stream>>>(U, V, outbuf, hbase, pbuf, bar);
}